// graph_89300960018517
// MI455X (gfx1250) — compile-verified
//
#include <hip/hip_runtime.h>
#include <cstdint>
#include <cstddef>

// Problem constants (match the reference generator exactly).
#define BB 4
#define CC 32
#define NN 50000
#define EE 800000

// ---------------------------------------------------------------------------
// Kernel 1: transpose x (B,C,N) -> xT (B,N,C), 32x32 LDS tiles.
// Full tiles are staged into LDS with the gfx1250 async global->LDS copy
// (ASYNCcnt path), tail tiles use guarded plain loads.
// ---------------------------------------------------------------------------
__global__ __launch_bounds__(256) void k_transpose_x(const float* __restrict__ x,
                                                     float* __restrict__ xT,
                                                     int nper) {
  __shared__ float tile[CC][CC + 1];  // +1 pad: no LDS bank conflicts
  const int b  = blockIdx.y;
  const int n0 = blockIdx.x * CC;
  const float* xb  = x  + (size_t)b * CC * nper;
  float*       xTb = xT + (size_t)b * nper * CC;
  const int tx = threadIdx.x;  // 0..31 (lane within row)
  const int ty = threadIdx.y;  // 0..7

  if (n0 + CC <= nper) {
    // Full tile: async copy global -> LDS, per-lane addresses (CDNA5 path).
#pragma unroll
    for (int k = 0; k < 4; ++k) {
      const int c = ty + 8 * k;
      const unsigned goff = (unsigned)(((size_t)c * (size_t)nper + (size_t)(n0 + tx)) *
                                       sizeof(float));
      const unsigned lds = (unsigned)(uintptr_t)(&tile[c][tx]);  // low 32b = LDS offset
      asm volatile("global_load_async_to_lds_b32 %0, %1, %2 offset:0"
                   :
                   : "v"(lds), "v"(goff), "s"(xb)
                   : "memory");
    }
    asm volatile("s_wait_asynccnt 0" ::: "memory");
  } else {
    // Tail tile: guarded scalar loads.
#pragma unroll
    for (int k = 0; k < 4; ++k) {
      const int c = ty + 8 * k;
      const int n = n0 + tx;
      tile[c][tx] = (n < nper) ? xb[(size_t)c * nper + n] : 0.0f;
    }
  }
  __syncthreads();

#pragma unroll
  for (int k = 0; k < 4; ++k) {
    const int n = n0 + ty + 8 * k;
    if (n < nper) xTb[(size_t)n * CC + tx] = tile[tx][ty + 8 * k];
  }
}

// ---------------------------------------------------------------------------
// Kernel 2: zero a float buffer (float4 stores).
// ---------------------------------------------------------------------------
__global__ __launch_bounds__(256) void k_zero4(float4* __restrict__ p, long long n4) {
  long long idx = (long long)blockIdx.x * blockDim.x + threadIdx.x;
  const long long stride = (long long)gridDim.x * blockDim.x;
  const float4 z = make_float4(0.f, 0.f, 0.f, 0.f);
  for (long long t = idx; t < n4; t += stride) p[t] = z;
}

// ---------------------------------------------------------------------------
// Kernel 3: edge scatter in transposed (B,N,C) layout.
// FOUR threads per edge (8 channels each): per-thread work is 4 b128 gathers
// + 16 line-local no-return f32 atomics -> short dependency chains, high
// wave count to hide L2 atomic latency. Adjacent lanes (quarters of one
// edge) touch consecutive 32B chunks of the same cachelines. Edge streams
// (W, i, j) are single-use -> non-temporal loads so L2 stays reserved for
// xT / outT (both fit in the 192MB L2).
// ---------------------------------------------------------------------------
__global__ __launch_bounds__(256) void k_edge_scatter(const float* __restrict__ xT,
                                                      const float* __restrict__ W,
                                                      const int* __restrict__ iI,
                                                      const int* __restrict__ jI,
                                                      float* __restrict__ outT,
                                                      int nper, int ne) {
  const int b = blockIdx.y;
  const long long t = (long long)blockIdx.x * blockDim.x + threadIdx.x;
  const int e = (int)(t >> 2);   // edge index
  const int q = (int)(t & 3);    // quarter: which 8 channels
  if (e >= ne) return;

  const float* xb = xT   + (size_t)b * nper * CC;
  float*       ob = outT + (size_t)b * nper * CC;

  const float w = __builtin_nontemporal_load(W  + (size_t)b * ne + e);
  const int   i = __builtin_nontemporal_load(iI + (size_t)b * ne + e);
  const int   j = __builtin_nontemporal_load(jI + (size_t)b * ne + e);
  const float d = w * w;

  const float4* xi = (const float4*)(xb + (size_t)i * CC) + 2 * q;  // 32B aligned
  const float4* xj = (const float4*)(xb + (size_t)j * CC) + 2 * q;
  float* oi = ob + (size_t)i * CC + 8 * q;
  float* oj = ob + (size_t)j * CC + 8 * q;

  const float4 a0 = xi[0];
  const float4 a1 = xi[1];
  const float4 c0 = xj[0];
  const float4 c1 = xj[1];

  const float g0 = d * (c0.x - a0.x);
  const float g1 = d * (c0.y - a0.y);
  const float g2 = d * (c0.z - a0.z);
  const float g3 = d * (c0.w - a0.w);
  const float g4 = d * (c1.x - a1.x);
  const float g5 = d * (c1.y - a1.y);
  const float g6 = d * (c1.z - a1.z);
  const float g7 = d * (c1.w - a1.w);

  atomicAdd(oi + 0, g0);
  atomicAdd(oi + 1, g1);
  atomicAdd(oi + 2, g2);
  atomicAdd(oi + 3, g3);
  atomicAdd(oi + 4, g4);
  atomicAdd(oi + 5, g5);
  atomicAdd(oi + 6, g6);
  atomicAdd(oi + 7, g7);
  atomicAdd(oj + 0, -g0);
  atomicAdd(oj + 1, -g1);
  atomicAdd(oj + 2, -g2);
  atomicAdd(oj + 3, -g3);
  atomicAdd(oj + 4, -g4);
  atomicAdd(oj + 5, -g5);
  atomicAdd(oj + 6, -g6);
  atomicAdd(oj + 7, -g7);
}

// ---------------------------------------------------------------------------
// Kernel 4: transpose outT (B,N,C) -> out (B,C,N), 32x32 LDS tiles.
// Full tiles staged via async global->LDS copy as well.
// ---------------------------------------------------------------------------
__global__ __launch_bounds__(256) void k_transpose_out(const float* __restrict__ outT,
                                                       float* __restrict__ out,
                                                       int nper) {
  __shared__ float tile[CC][CC + 1];
  const int b  = blockIdx.y;
  const int n0 = blockIdx.x * CC;
  const float* ob   = outT + (size_t)b * nper * CC;
  float*       outb = out  + (size_t)b * CC * nper;
  const int tx = threadIdx.x;
  const int ty = threadIdx.y;

  if (n0 + CC <= nper) {
#pragma unroll
    for (int k = 0; k < 4; ++k) {
      const int n = n0 + ty + 8 * k;
      const unsigned goff = (unsigned)(((size_t)n * CC + (size_t)tx) * sizeof(float));
      const unsigned lds = (unsigned)(uintptr_t)(&tile[ty + 8 * k][tx]);
      asm volatile("global_load_async_to_lds_b32 %0, %1, %2 offset:0"
                   :
                   : "v"(lds), "v"(goff), "s"(ob)
                   : "memory");
    }
    asm volatile("s_wait_asynccnt 0" ::: "memory");
  } else {
#pragma unroll
    for (int k = 0; k < 4; ++k) {
      const int n = n0 + ty + 8 * k;
      if (n < nper) tile[ty + 8 * k][tx] = ob[(size_t)n * CC + tx];
    }
  }
  __syncthreads();

#pragma unroll
  for (int k = 0; k < 4; ++k) {
    const int c = ty + 8 * k;
    const int n = n0 + tx;
    if (n < nper) outb[(size_t)c * nper + n] = tile[tx][c];
  }
}

// ---------------------------------------------------------------------------
// Fallback: direct-layout scatter (used only if workspace is too small).
// ---------------------------------------------------------------------------
__global__ __launch_bounds__(256) void k_edge_scatter_direct(const float* __restrict__ x,
                                                             const float* __restrict__ W,
                                                             const int* __restrict__ iI,
                                                             const int* __restrict__ jI,
                                                             float* __restrict__ out,
                                                             int nper, int ne) {
  const int b = blockIdx.y;
  const int e = blockIdx.x * blockDim.x + threadIdx.x;
  if (e >= ne) return;
  const float* xb = x   + (size_t)b * CC * nper;
  float*       ob = out + (size_t)b * CC * nper;
  const float w = __builtin_nontemporal_load(W  + (size_t)b * ne + e);
  const int   i = __builtin_nontemporal_load(iI + (size_t)b * ne + e);
  const int   j = __builtin_nontemporal_load(jI + (size_t)b * ne + e);
  const float d = w * w;
#pragma unroll 8
  for (int c = 0; c < CC; ++c) {
    const float g = d * (xb[(size_t)c * nper + j] - xb[(size_t)c * nper + i]);
    atomicAdd(ob + (size_t)c * nper + i,  g);
    atomicAdd(ob + (size_t)c * nper + j, -g);
  }
}

extern "C" void kernel_launch(void* const* d_in, const int* in_sizes, int n_in,
                              void* d_out, int out_size, void* d_ws, size_t ws_size,
                              hipStream_t stream) {
  (void)in_sizes; (void)n_in; (void)out_size;
  const float* x  = (const float*)d_in[0];  // (B,C,N) f32
  const float* W  = (const float*)d_in[1];  // (B,E)   f32
  const int*   iI = (const int*)d_in[2];    // (B,E)   i32
  const int*   jI = (const int*)d_in[3];    // (B,E)   i32
  float* out = (float*)d_out;               // (B,C,N) f32

  const int Bn = BB, Nn = NN, En = EE;
  const size_t plane = (size_t)Bn * (size_t)Nn * (size_t)CC;  // == B*C*N elements
  const size_t need  = 2 * plane * sizeof(float);             // xT + outT

  const dim3 tb(32, 8);
  const dim3 tg((Nn + CC - 1) / CC, Bn);

  if (ws_size >= need) {
    float* xT   = (float*)d_ws;
    float* outT = xT + plane;
    const long long threads = 4LL * En;  // 4 threads per edge
    const dim3 eb(256);
    const dim3 eg((unsigned)((threads + 255) / 256), Bn);
    k_transpose_x<<<tg, tb, 0, stream>>>(x, xT, Nn);
    k_zero4<<<2048, 256, 0, stream>>>((float4*)outT, (long long)(plane / 4));
    k_edge_scatter<<<eg, eb, 0, stream>>>(xT, W, iI, jI, outT, Nn, En);
    k_transpose_out<<<tg, tb, 0, stream>>>(outT, out, Nn);
  } else {
    const dim3 eb(256);
    const dim3 eg((En + 255) / 256, Bn);
    k_zero4<<<2048, 256, 0, stream>>>((float4*)out, (long long)(plane / 4));
    k_edge_scatter_direct<<<eg, eb, 0, stream>>>(x, W, iI, jI, out, Nn, En);
  }
}